// GeometricStructureEmbedding_21139829030943
// MI455X (gfx1250) — compile-verified
//
#include <hip/hip_runtime.h>
#include <hip/hip_bf16.h>
#include <math.h>

// Problem constants (fixed by the reference setup_inputs)
#define BATCH   2
#define NPTS    1024
#define HID     128
#define HALFH   64
#define KNN     10
#define NBINS   256
#define FACTOR_A 3.8197186342054885f      /* 180/(15*pi) = 12/pi */
#define NEG_LN1E4_OVER_64 (-0.14391156831212787f)

typedef float v2f __attribute__((ext_vector_type(2)));
typedef float v8f __attribute__((ext_vector_type(8)));

// ---------------------------------------------------------------------------
// Kernel A: build sinusoid lookup table row c (in LDS) and project it:
//   TdW[c][h] = (sum_k tab[k] * Wd[h][k]) / NPTS      (mean folded in)
//   TaW[c][h] =  sum_k tab[k] * Wa[h][k] + ba[h]
// grid = 256 blocks (one per bin value c), 128 threads (one per channel h)
// ---------------------------------------------------------------------------
__global__ __launch_bounds__(HID)
void gse_tables_kernel(const float* __restrict__ Wd, const float* __restrict__ bd_unused,
                       const float* __restrict__ Wa, const float* __restrict__ ba,
                       float* __restrict__ TdW, float* __restrict__ TaW) {
    __shared__ float tab[HID];
    const int c = blockIdx.x;
    const int t = threadIdx.x;

    // sinusoid: even channel 2m -> sin(c*dt[m]), odd 2m+1 -> cos(c*dt[m])
    {
        const int m = t >> 1;
        const float dt = __expf(0.0f) * expf((float)m * NEG_LN1E4_OVER_64);
        const float arg = (float)c * dt;
        tab[t] = (t & 1) ? cosf(arg) : sinf(arg);
    }
    __syncthreads();

    float sd = 0.0f, sa = 0.0f;
    const float* wdr = Wd + t * HID;
    const float* war = Wa + t * HID;
    for (int k = 0; k < HID; ++k) {
        const float tv = tab[k];
        sd = fmaf(tv, wdr[k], sd);
        sa = fmaf(tv, war[k], sa);
    }
    TdW[c * HID + t] = sd * (1.0f / (float)NPTS);
    TaW[c * HID + t] = sa + ba[t];
}

// ---------------------------------------------------------------------------
// Kernel B: one block per point (B*N blocks, 128 threads).
//  - distances to all N points, 256-bin histogram (LDS atomics) -> histF
//  - stable top-10 nearest (argmin with lower-index tie-break, matches top_k)
//  - unit vectors, 100 pairwise angle bins
//  - embed_a[i][h] = max over 100 pairs of TaW[bin][h]  -> embA
// ---------------------------------------------------------------------------
__global__ __launch_bounds__(128)
void gse_point_kernel(const float* __restrict__ points,
                      const float* __restrict__ TaW,
                      float* __restrict__ histF,
                      float* __restrict__ embA) {
    __shared__ float        dist[NPTS];
    __shared__ unsigned int hist[NBINS];
    __shared__ float        rv[128];
    __shared__ int          ri[128];
    __shared__ int          nn[KNN];
    __shared__ float        ux[KNN], uy[KNN], uz[KNN];
    __shared__ int          idxA[KNN * KNN];

    const int bid = blockIdx.x;             // 0 .. B*N-1
    const int b   = bid >> 10;              // NPTS == 1024
    const int i   = bid & (NPTS - 1);
    const int t   = threadIdx.x;

    const float* P = points + (size_t)b * NPTS * 3;
    const float pix = P[i * 3 + 0];
    const float piy = P[i * 3 + 1];
    const float piz = P[i * 3 + 2];

    // zero histogram
    hist[t] = 0u;
    hist[t + 128] = 0u;
    __syncthreads();

    // distance + binning pass
    for (int j = t; j < NPTS; j += 128) {
        const float dx = pix - P[j * 3 + 0];
        const float dy = piy - P[j * 3 + 1];
        const float dz = piz - P[j * 3 + 2];
        const float d  = sqrtf(dx * dx + dy * dy + dz * dz);
        dist[j] = d;
        int bin = (int)rintf(d / 0.2f);     // SIGMA_D, banker's rounding like jnp.round
        bin = bin < 0 ? 0 : (bin > 255 ? 255 : bin);
        atomicAdd(&hist[bin], 1u);
    }
    __syncthreads();

    // export histogram as float GEMM A-row
    {
        float* hrow = histF + (size_t)bid * NBINS;
        hrow[t]       = (float)hist[t];
        hrow[t + 128] = (float)hist[t + 128];
    }

    // stable top-K argmin (includes self at distance 0)
    for (int k = 0; k < KNN; ++k) {
        float best = INFINITY; int bi = NPTS;
        for (int j = t; j < NPTS; j += 128) {
            const float d = dist[j];
            if (d < best || (d == best && j < bi)) { best = d; bi = j; }
        }
        rv[t] = best; ri[t] = bi;
        __syncthreads();
        if (t == 0) {
            float gb = INFINITY; int gi = NPTS;
            for (int w = 0; w < 128; ++w) {
                const float v = rv[w]; const int id = ri[w];
                if (v < gb || (v == gb && id < gi)) { gb = v; gi = id; }
            }
            nn[k] = gi;
            dist[gi] = INFINITY;            // exclude for next round
        }
        __syncthreads();
    }

    // unit vectors to the K neighbours
    if (t < KNN) {
        const int j = nn[t];
        const float vx = P[j * 3 + 0] - pix;
        const float vy = P[j * 3 + 1] - piy;
        const float vz = P[j * 3 + 2] - piz;
        const float nrm = sqrtf(vx * vx + vy * vy + vz * vz) + 1e-8f;
        ux[t] = vx / nrm; uy[t] = vy / nrm; uz[t] = vz / nrm;
    }
    __syncthreads();

    // angle bins for all K*K pairs
    if (t < KNN * KNN) {
        const int k = t / KNN, l = t % KNN;
        float cs = ux[k] * ux[l] + uy[k] * uy[l] + uz[k] * uz[l];
        cs = fminf(1.0f, fmaxf(-1.0f, cs));
        const float ang = acosf(cs);
        int ia = (int)rintf(ang * FACTOR_A);
        ia = ia < 0 ? 0 : (ia > 255 ? 255 : ia);
        idxA[t] = ia;
    }
    __syncthreads();

    // embed_a: channel-parallel max over the 100 projected rows
    {
        float acc = -INFINITY;
        for (int q = 0; q < KNN * KNN; ++q)
            acc = fmaxf(acc, TaW[idxA[q] * HID + t]);
        embA[(size_t)bid * HID + t] = acc;
    }
}

// ---------------------------------------------------------------------------
// Kernel C: WMMA f32 GEMM  out = histF(2048x256) @ TdW(256x128) + bd + embA
// One wave (32 threads) per 16x16 output tile, K-loop of 64 x v_wmma_f32_16x16x4_f32.
// Layouts per CDNA5 ISA 7.12.2:
//   A 16x4 : lane&15 = M, VGPR{0,1} = K {2*half, 2*half+1},  half = lane>>4
//   B 4x16 : lane&15 = N, VGPR{0,1} = K {2*half, 2*half+1}
//   C 16x16: lane&15 = N, VGPR r -> M = r + 8*half
// ---------------------------------------------------------------------------
__global__ __launch_bounds__(32)
void gse_gemm_kernel(const float* __restrict__ histF,
                     const float* __restrict__ TdW,
                     const float* __restrict__ embA,
                     const float* __restrict__ bd,
                     float* __restrict__ out) {
    const int tm   = blockIdx.x;            // 0..(B*N/16 - 1)
    const int tn   = blockIdx.y;            // 0..(HID/16 - 1)
    const int lane = threadIdx.x;
    const int half = lane >> 4;
    const int mn   = lane & 15;

    v8f c = {};
    const float* Arow = histF + (size_t)(tm * 16 + mn) * NBINS;
    const float* Bcol = TdW + tn * 16 + mn;

#if __has_builtin(__builtin_amdgcn_wmma_f32_16x16x4_f32)
    for (int kk = 0; kk < NBINS; kk += 4) {
        v2f a, bvec;
        a.x    = Arow[kk + 2 * half];
        a.y    = Arow[kk + 2 * half + 1];
        bvec.x = Bcol[(size_t)(kk + 2 * half) * HID];
        bvec.y = Bcol[(size_t)(kk + 2 * half + 1) * HID];
        c = __builtin_amdgcn_wmma_f32_16x16x4_f32(
                false, a, false, bvec, (short)0, c, false, false);
    }
#else
    // scalar fallback (should not be taken on gfx1250)
    for (int kk = 0; kk < NBINS; ++kk) {
        const float bv = Bcol[(size_t)kk * HID];
        for (int r = 0; r < 8; ++r) {
            const float av = histF[(size_t)(tm * 16 + r + 8 * half) * NBINS + kk];
            c[r] = fmaf(av, bv, c[r]);
        }
    }
#endif

    const int col = tn * 16 + mn;
    const float bb = bd[col];
    for (int r = 0; r < 8; ++r) {
        const int row = tm * 16 + r + 8 * half;
        out[(size_t)row * HID + col] = c[r] + bb + embA[(size_t)row * HID + col];
    }
}

// ---------------------------------------------------------------------------
extern "C" void kernel_launch(void* const* d_in, const int* in_sizes, int n_in,
                              void* d_out, int out_size, void* d_ws, size_t ws_size,
                              hipStream_t stream) {
    const float* points = (const float*)d_in[0];   // (B, N, 3)
    const float* Wd     = (const float*)d_in[1];   // (H, H)
    const float* bd     = (const float*)d_in[2];   // (H)
    const float* Wa     = (const float*)d_in[3];   // (H, H)
    const float* ba     = (const float*)d_in[4];   // (H)
    float*       outp   = (float*)d_out;           // (B, N, H)

    // workspace layout (floats)
    float* TdW   = (float*)d_ws;                    // 256*128
    float* TaW   = TdW + NBINS * HID;               // 256*128
    float* histF = TaW + NBINS * HID;               // 2048*256
    float* embA  = histF + (size_t)BATCH * NPTS * NBINS; // 2048*128

    // A: projected sinusoid tables
    gse_tables_kernel<<<NBINS, HID, 0, stream>>>(Wd, bd, Wa, ba, TdW, TaW);

    // B: per-point histogram + topK + angle gather-max
    gse_point_kernel<<<BATCH * NPTS, 128, 0, stream>>>(points, TaW, histF, embA);

    // C: WMMA GEMM + epilogue
    dim3 grid((BATCH * NPTS) / 16, HID / 16);
    gse_gemm_kernel<<<grid, 32, 0, stream>>>(histF, TdW, embA, bd, outp);
}